// EPrompt_69475390980437
// MI455X (gfx1250) — compile-verified
//
#include <hip/hip_runtime.h>

#define B_   256
#define N_   197
#define C_   768
#define P_   1024
#define L_   6
#define LP_  8
#define H_   12
#define DH_  64
#define TOPK 4
#define CHUNK (LP_*H_*DH_)                      /* 6144 floats = 24 KB */
#define OUT_PROMPT (L_*B_*2*TOPK*LP_*H_*DH_)    /* 75,497,472 */

typedef __attribute__((ext_vector_type(2))) float v2f;
typedef __attribute__((ext_vector_type(4))) float v4f;
typedef __attribute__((ext_vector_type(8))) float v8f;

// ---------------------------------------------------------------------------
// 1) x_mean over N, then L2-normalize -> x_n  (one block per batch row)
// ---------------------------------------------------------------------------
__global__ __launch_bounds__(256)
void k_xmean_norm(const float* __restrict__ x, float* __restrict__ xn) {
    const int b = blockIdx.x, t = threadIdx.x;
    const float* base = x + (size_t)b * N_ * C_;
    float a0 = 0.f, a1 = 0.f, a2 = 0.f;
    for (int n = 0; n < N_; ++n) {
        const float* row = base + n * C_;
        a0 += row[t]; a1 += row[t + 256]; a2 += row[t + 512];
    }
    const float inv = 1.0f / (float)N_;
    a0 *= inv; a1 *= inv; a2 *= inv;
    __shared__ float red[256];
    red[t] = a0 * a0 + a1 * a1 + a2 * a2;
    __syncthreads();
    for (int s = 128; s > 0; s >>= 1) { if (t < s) red[t] += red[t + s]; __syncthreads(); }
    const float sc = rsqrtf(fmaxf(red[0], 1e-12f));
    float* o = xn + (size_t)b * C_;
    o[t] = a0 * sc; o[t + 256] = a1 * sc; o[t + 512] = a2 * sc;
}

// ---------------------------------------------------------------------------
// 2) L2-normalize prompt_key -> key_n  (one block per prompt row)
// ---------------------------------------------------------------------------
__global__ __launch_bounds__(256)
void k_key_norm(const float* __restrict__ pk, float* __restrict__ kn) {
    const int p = blockIdx.x, t = threadIdx.x;
    const float* row = pk + (size_t)p * C_;
    const float a0 = row[t], a1 = row[t + 256], a2 = row[t + 512];
    __shared__ float red[256];
    red[t] = a0 * a0 + a1 * a1 + a2 * a2;
    __syncthreads();
    for (int s = 128; s > 0; s >>= 1) { if (t < s) red[t] += red[t + s]; __syncthreads(); }
    const float sc = rsqrtf(fmaxf(red[0], 1e-12f));
    float* o = kn + (size_t)p * C_;
    o[t] = a0 * sc; o[t + 256] = a1 * sc; o[t + 512] = a2 * sc;
}

// ---------------------------------------------------------------------------
// 3) similarity = x_n @ key_n^T via V_WMMA_F32_16X16X4_F32.
//    One wave32 per 16x16 output tile; K=768 in steps of 4.
//    A 16x4 layout: lanes 0-15 hold {K, K+1}, lanes 16-31 hold {K+2, K+3}.
//    B 4x16 mirrored. C/D: VGPR r -> rows r (lanes 0-15) / r+8 (lanes 16-31).
// ---------------------------------------------------------------------------
__global__ __launch_bounds__(32)
void k_sim_wmma(const float* __restrict__ xn, const float* __restrict__ kn,
                float* __restrict__ sim) {
    const int tile = blockIdx.x;       // 0..1023
    const int pt   = tile & 63;        // 64 tiles across P
    const int bt   = tile >> 6;        // 16 tiles across B
    const int lane = threadIdx.x;      // 0..31
    const int half = lane >> 4;        // 0 or 1
    const int l16  = lane & 15;

    const float* arow = xn + (size_t)(bt * 16 + l16) * C_ + half * 2;
    const float* brow = kn + (size_t)(pt * 16 + l16) * C_ + half * 2;

    v8f acc = {};
    for (int c = 0; c < C_; c += 4) {
        v2f a, b;
        a.x = arow[c]; a.y = arow[c + 1];
        b.x = brow[c]; b.y = brow[c + 1];
        acc = __builtin_amdgcn_wmma_f32_16x16x4_f32(
            /*neg_a=*/false, a, /*neg_b=*/false, b,
            /*c_mod=*/(short)0, acc, /*reuse_a=*/false, /*reuse_b=*/false);
    }

    float* out = sim + (size_t)(bt * 16 + half * 8) * P_ + pt * 16 + l16;
#pragma unroll
    for (int r = 0; r < 8; ++r) out[(size_t)r * P_] = acc[r];
}

// ---------------------------------------------------------------------------
// 4) top-4 per row (one thread per batch row; strict > keeps lowest index on
//    ties, matching jax.lax.top_k).
// ---------------------------------------------------------------------------
__global__ __launch_bounds__(32)
void k_topk(const float* __restrict__ sim, int* __restrict__ idx) {
    const int b = blockIdx.x * blockDim.x + threadIdx.x;
    if (b >= B_) return;
    const float* row = sim + (size_t)b * P_;
    float v0 = -INFINITY, v1 = -INFINITY, v2 = -INFINITY, v3 = -INFINITY;
    int   i0 = 0, i1 = 0, i2 = 0, i3 = 0;
    for (int p = 0; p < P_; ++p) {
        const float v = row[p];
        if (v > v3) {
            if (v > v0)      { v3=v2;i3=i2; v2=v1;i2=i1; v1=v0;i1=i0; v0=v;i0=p; }
            else if (v > v1) { v3=v2;i3=i2; v2=v1;i2=i1; v1=v;i1=p; }
            else if (v > v2) { v3=v2;i3=i2; v2=v;i2=p; }
            else             { v3=v;i3=p; }
        }
    }
    idx[b*4+0]=i0; idx[b*4+1]=i1; idx[b*4+2]=i2; idx[b*4+3]=i3;
}

// ---------------------------------------------------------------------------
// 5) The big gather + transpose-fold. One block per contiguous 24KB chunk:
//    out[l][b][d][k*8 + lp][h][dh] = prompt[l][d][idx[b][k]][lp][h][dh]
//    302 MB streamed out -> non-temporal stores (output > L2); loads RT
//    (repeated prompt indices across batch hit L2).
// ---------------------------------------------------------------------------
__global__ __launch_bounds__(256)
void k_gather(const float* __restrict__ prompt, const int* __restrict__ idx,
              float* __restrict__ out) {
    const int chunk = blockIdx.x;        // 12288 chunks = l*2048 + b*8 + d*4 + k
    const int k = chunk & 3;
    const int d = (chunk >> 2) & 1;
    const int b = (chunk >> 3) & 255;
    const int l = chunk >> 11;
    const int p = idx[b * 4 + k];

    const v4f* __restrict__ src =
        (const v4f*)(prompt + ((size_t)(l * 2 + d) * P_ + p) * CHUNK);
    v4f* __restrict__ dst =
        (v4f*)(out + ((size_t)((l * B_ + b) * 2 + d) * (TOPK * LP_) + k * LP_) * (H_ * DH_));

    const int t = threadIdx.x;
#pragma unroll
    for (int j = 0; j < 6; ++j) {               // 6*256*16B = 24KB
        const v4f v = src[j * 256 + t];
        __builtin_nontemporal_store(v, dst + j * 256 + t);
    }
}

// ---------------------------------------------------------------------------
// 6) reduce_sim = sum_b sum_k dot(key_n[idx[b,k]], x_n[b]) / B
//    Two-stage deterministic tree reduction (no float atomics).
// ---------------------------------------------------------------------------
__global__ __launch_bounds__(256)
void k_rsim_partial(const float* __restrict__ xn, const float* __restrict__ kn,
                    const int* __restrict__ idx, float* __restrict__ part) {
    const int b = blockIdx.x, t = threadIdx.x;
    const float* xr = xn + (size_t)b * C_;
    float acc = 0.f;
    for (int k = 0; k < TOPK; ++k) {
        const float* kr = kn + (size_t)idx[b * 4 + k] * C_;
        acc += xr[t] * kr[t] + xr[t + 256] * kr[t + 256] + xr[t + 512] * kr[t + 512];
    }
    __shared__ float red[256];
    red[t] = acc;
    __syncthreads();
    for (int s = 128; s > 0; s >>= 1) { if (t < s) red[t] += red[t + s]; __syncthreads(); }
    if (t == 0) part[b] = red[0];
}

__global__ __launch_bounds__(256)
void k_rsim_final(const float* __restrict__ part, float* __restrict__ out) {
    const int t = threadIdx.x;
    __shared__ float red[256];
    red[t] = part[t];
    __syncthreads();
    for (int s = 128; s > 0; s >>= 1) { if (t < s) red[t] += red[t + s]; __syncthreads(); }
    if (t == 0) out[OUT_PROMPT] = red[0] / (float)B_;
}

// ---------------------------------------------------------------------------
extern "C" void kernel_launch(void* const* d_in, const int* in_sizes, int n_in,
                              void* d_out, int out_size, void* d_ws, size_t ws_size,
                              hipStream_t stream) {
    const float* x_embed    = (const float*)d_in[0];   // (256,197,768)
    const float* prompt     = (const float*)d_in[1];   // (6,2,1024,8,12,64)
    const float* prompt_key = (const float*)d_in[2];   // (1024,768)
    float* out = (float*)d_out;                        // 75,497,472 + 1 floats

    char* ws = (char*)d_ws;
    float* xn   = (float*)(ws);                                         // 768 KB
    float* kn   = (float*)(ws + 786432);                                // 3 MB
    float* sim  = (float*)(ws + 786432 + 3145728);                      // 1 MB
    int*   idx  = (int*)  (ws + 786432 + 3145728 + 1048576);            // 4 KB
    float* part = (float*)(ws + 786432 + 3145728 + 1048576 + 4096);     // 1 KB

    k_xmean_norm  <<<B_,    256, 0, stream>>>(x_embed, xn);
    k_key_norm    <<<P_,    256, 0, stream>>>(prompt_key, kn);
    k_sim_wmma    <<<1024,   32, 0, stream>>>(xn, kn, sim);
    k_topk        <<<8,      32, 0, stream>>>(sim, idx);
    k_gather      <<<12288, 256, 0, stream>>>(prompt, idx, out);
    k_rsim_partial<<<B_,    256, 0, stream>>>(xn, kn, idx, part);
    k_rsim_final  <<<1,     256, 0, stream>>>(part, out);
}